// DyGraphConv2d_45320494907723
// MI455X (gfx1250) — compile-verified
//
#include <hip/hip_runtime.h>
#include <hip/hip_bf16.h>
#include <math.h>

// ---- problem constants (fixed by reference) ----
#define BB   16
#define CC   192
#define HH   28
#define WW   28
#define NN   784          // H*W
#define KNN_ 9
#define COUT 384
#define CR   12           // C / r
#define CTOT 768          // 2*COUT stacked rows (W1-W2 ; W2)
#define NT   49           // NN / 16
#define MT_G 49           // gram M tiles
#define MT_E 48           // CTOT / 16

typedef float v2f __attribute__((ext_vector_type(2)));
typedef float v8f __attribute__((ext_vector_type(8)));
typedef unsigned int u32x4 __attribute__((ext_vector_type(4)));
typedef int i32x4 __attribute__((ext_vector_type(4)));
typedef int i32x8 __attribute__((ext_vector_type(8)));

__device__ __forceinline__ float sigmoidf_(float z) { return 1.0f / (1.0f + expf(-z)); }

// ---------------- K1: per-(b,c) mean & max over N ----------------
__global__ void k_chan_pool(const float* __restrict__ x, float* pmean, float* pmax) {
    int bc = blockIdx.x;                       // b*CC + c
    const float* row = x + (size_t)bc * NN;
    float s = 0.f, m = -INFINITY;
    for (int n = threadIdx.x; n < NN; n += blockDim.x) { float v = row[n]; s += v; m = fmaxf(m, v); }
    __shared__ float ss[256], sm[256];
    ss[threadIdx.x] = s; sm[threadIdx.x] = m; __syncthreads();
    for (int o = 128; o > 0; o >>= 1) {
        if (threadIdx.x < o) { ss[threadIdx.x] += ss[threadIdx.x + o]; sm[threadIdx.x] = fmaxf(sm[threadIdx.x], sm[threadIdx.x + o]); }
        __syncthreads();
    }
    if (threadIdx.x == 0) { pmean[bc] = ss[0] / (float)NN; pmax[bc] = sm[0]; }
}

// ---------------- K2: channel MLP + sigmoid (tiny) ----------------
__global__ void k_chan_mlp(const float* pmean, const float* pmax,
                           const float* __restrict__ w1, const float* __restrict__ b1,
                           const float* __restrict__ w2, const float* __restrict__ b2,
                           float* ch) {
    int b = blockIdx.x, t = threadIdx.x;       // 192 threads
    __shared__ float vm[CC], vx[CC], hm[CR], hx[CR];
    vm[t] = pmean[b * CC + t]; vx[t] = pmax[b * CC + t];
    __syncthreads();
    if (t < CR) {
        float am = b1[t], ax = b1[t];
        for (int c = 0; c < CC; ++c) { float w = w1[t * CC + c]; am += w * vm[c]; ax += w * vx[c]; }
        hm[t] = fmaxf(am, 0.f); hx[t] = fmaxf(ax, 0.f);
    }
    __syncthreads();
    float om = b2[t], ox = b2[t];
    for (int j = 0; j < CR; ++j) { float w = w2[t * CR + j]; om += w * hm[j]; ox += w * hx[j]; }
    ch[b * CC + t] = sigmoidf_(om + ox);
}

// ---------------- K3a: spatial mean/max over channels of x*ch ----------------
__global__ void k_sp_in(const float* __restrict__ x, const float* __restrict__ ch, float* sp_in) {
    int t = blockIdx.x * blockDim.x + threadIdx.x;
    if (t >= BB * NN) return;
    int b = t / NN, n = t % NN;
    float s = 0.f, m = -INFINITY;
    for (int c = 0; c < CC; ++c) {
        float v = x[((size_t)(b * CC + c)) * NN + n] * ch[b * CC + c];
        s += v; m = fmaxf(m, v);
    }
    sp_in[(size_t)b * 2 * NN + n]       = s / (float)CC;
    sp_in[(size_t)b * 2 * NN + NN + n]  = m;
}

// ---------------- K3b: 7x7 2-in 1-out conv, pad 3, + sigmoid ----------------
__global__ void k_sp_conv(const float* __restrict__ sp_in, const float* __restrict__ w_sp,
                          const float* __restrict__ b_sp, float* sp_att) {
    int t = blockIdx.x * blockDim.x + threadIdx.x;
    if (t >= BB * NN) return;
    int b = t / NN, n = t % NN, h = n / WW, w = n % WW;
    float acc = b_sp[0];
    for (int i = 0; i < 2; ++i)
        for (int dy = 0; dy < 7; ++dy) {
            int hh = h + dy - 3; if (hh < 0 || hh >= HH) continue;
            for (int dx = 0; dx < 7; ++dx) {
                int ww2 = w + dx - 3; if (ww2 < 0 || ww2 >= WW) continue;
                acc += sp_in[(size_t)b * 2 * NN + i * NN + hh * WW + ww2] * w_sp[(i * 7 + dy) * 7 + dx];
            }
        }
    sp_att[b * NN + n] = sigmoidf_(acc);
}

// ---------------- K4: per-(b,c) softmax-based attention, xf_att = x * att ----------------
__global__ void k_att_row(const float* __restrict__ x, const float* __restrict__ ch,
                          const float* __restrict__ sp_att, float* __restrict__ xf_att) {
    int bc = blockIdx.x, b = bc / CC, t = threadIdx.x;
    const float* xrow = x + (size_t)bc * NN;
    const float* sprow = sp_att + b * NN;
    float chv = ch[bc];
    float s[4], e[4];
    float lm = -INFINITY;
    for (int i = 0; i < 4; ++i) {
        int n = t + i * 256;
        if (n < NN) { s[i] = xrow[n] * chv * sprow[n]; lm = fmaxf(lm, s[i]); }
        else s[i] = -INFINITY;
    }
    __shared__ float red[256];
    red[t] = lm; __syncthreads();
    for (int o = 128; o > 0; o >>= 1) { if (t < o) red[t] = fmaxf(red[t], red[t + o]); __syncthreads(); }
    float smax = red[0]; __syncthreads();
    float ls = 0.f;
    for (int i = 0; i < 4; ++i) { int n = t + i * 256; if (n < NN) { e[i] = expf(s[i] - smax); ls += e[i]; } }
    red[t] = ls; __syncthreads();
    for (int o = 128; o > 0; o >>= 1) { if (t < o) red[t] += red[t + o]; __syncthreads(); }
    float Z = red[0];
    float amax = 1.0f / Z;                      // max of softmax = exp(0)/Z
    for (int i = 0; i < 4; ++i) {
        int n = t + i * 256;
        if (n < NN) {
            float a = e[i] / Z;
            float an = a / (amax + 1e-10f);
            float att = (2.f * an - 1.f) / 40.f + 1.f;   // SCALAR = 20
            xf_att[(size_t)bc * NN + n] = xrow[n] * att;
        }
    }
}

// ---------------- K5: per-point channel normalization -> xnT (B,N,C), sq ----------------
__global__ void k_norm(const float* __restrict__ x, float* __restrict__ xnT, float* __restrict__ sq) {
    int bn = blockIdx.x, b = bn / NN, n = bn % NN, t = threadIdx.x;   // 256 threads, 192 active
    float v = 0.f;
    if (t < CC) v = x[((size_t)(b * CC + t)) * NN + n];
    __shared__ float red[256];
    red[t] = v * v; __syncthreads();
    for (int o = 128; o > 0; o >>= 1) { if (t < o) red[t] += red[t + o]; __syncthreads(); }
    float nrm = sqrtf(red[0]);
    float inv = 1.0f / fmaxf(nrm, 1e-12f);
    if (t < CC) xnT[(size_t)bn * CC + t] = v * inv;
    if (t == 0) sq[bn] = red[0] * inv * inv;
}

// ---------------- K6: Gram matrix G = Xn^T Xn via f32 WMMA (one wave per 16x16 tile) ----------------
__global__ void k_gram_wmma(const float* __restrict__ xnT, float* __restrict__ G) {
    int blk = blockIdx.x;
    int b = blk / (MT_G * NT); int rem = blk % (MT_G * NT);
    int ti = rem / NT, tj = rem % NT;
    int n0 = ti * 16, m0 = tj * 16;
    int lane = threadIdx.x;
    int r = lane & 15;
    int koff = (lane >> 4) * 2;                 // A/B frag layout: lane<16 -> K 0,1 ; lane>=16 -> K 2,3
    const float* abase = xnT + ((size_t)(b * NN + n0 + r)) * CC + koff;
    const float* bbase = xnT + ((size_t)(b * NN + m0 + r)) * CC + koff;
    v8f acc = {0.f, 0.f, 0.f, 0.f, 0.f, 0.f, 0.f, 0.f};
    for (int k = 0; k < CC; k += 4) {
        v2f a;  a.x  = abase[k]; a.y  = abase[k + 1];
        v2f bv; bv.x = bbase[k]; bv.y = bbase[k + 1];
        acc = __builtin_amdgcn_wmma_f32_16x16x4_f32(false, a, false, bv, (short)0, acc, false, false);
    }
    int rowoff = 8 * (lane >> 4);               // D layout: VGPR v -> row v (+8 for upper lanes)
    float* gout = G + ((size_t)(b * NN + n0 + rowoff)) * NN + m0 + r;
#pragma unroll
    for (int v = 0; v < 8; ++v) gout[(size_t)v * NN] = acc[v];
}

// ---------------- K7: 9-NN selection per point (strict-less insertion = stable ties) ----------------
__global__ void k_topk(const float* __restrict__ G, const float* __restrict__ sq, int* __restrict__ idx) {
    int t = blockIdx.x * blockDim.x + threadIdx.x;
    if (t >= BB * NN) return;
    int b = t / NN, n = t % NN;
    const float* grow = G + ((size_t)(b * NN + n)) * NN;
    const float* sqb = sq + b * NN;
    float sn = sqb[n];
    float bd[KNN_]; int bi[KNN_];
    for (int j = 0; j < KNN_; ++j) { bd[j] = INFINITY; bi[j] = 0; }
    for (int m = 0; m < NN; ++m) {
        float d = sn + sqb[m] - 2.0f * grow[m];
        if (d < bd[KNN_ - 1]) {
            int j = KNN_ - 1;
            while (j > 0 && bd[j - 1] > d) { bd[j] = bd[j - 1]; bi[j] = bi[j - 1]; --j; }
            bd[j] = d; bi[j] = m;
        }
    }
    for (int j = 0; j < KNN_; ++j) idx[(size_t)t * KNN_ + j] = bi[j];
}

// ---------------- K8a: build stacked weight Wc = [W1 - W2 ; W2] ----------------
__global__ void k_wc(const float* __restrict__ w_edge, float* __restrict__ Wc) {
    int t = blockIdx.x * blockDim.x + threadIdx.x;
    if (t >= CTOT * CC) return;
    int o = t / CC, c = t % CC;
    if (o < COUT) Wc[t] = w_edge[o * (2 * CC) + c] - w_edge[o * (2 * CC) + CC + c];
    else          Wc[t] = w_edge[(o - COUT) * (2 * CC) + CC + c];
}

// ---------------- K8b: AB = Wc (768x192) * xf_att (192x784) per batch ----------------
// One block per (b, n-tile). The 192x16 B-panel (12 KB) is DMA'd into LDS once by the
// Tensor Data Mover (D# built per ISA cdna5 8.3-8.4), then 8 waves x 6 M-tiles share it.
// This toolchain declares the 6-arg builtin: (u32x4 g0, i32x8 g1, i32x4 g2, i32x4 g3, i32x8, i32 cpol).
__global__ void __launch_bounds__(256) k_edge_wmma(const float* __restrict__ Wc,
                                                   const float* __restrict__ xf_att,
                                                   float* __restrict__ AB) {
    __shared__ float panel[CC * 16];            // [c][j], row = 16 floats -> conflict-free ds reads
    int blk = blockIdx.x;
    int b = blk / NT, tn = blk % NT;
    int n0 = tn * 16;
    const float* src = xf_att + (size_t)b * CC * NN + n0;   // tile start: row stride NN

#if __has_builtin(__builtin_amdgcn_tensor_load_to_lds)
    if (threadIdx.x < 32) {                     // wave 0 issues the TDM (EXEC ignored by TDM)
        unsigned long long ga = (unsigned long long)(size_t)src;
        unsigned lds_off = (unsigned)(size_t)(void*)panel;  // low 32 bits = LDS byte address
        u32x4 g0;
        g0[0] = 1u;                                           // count=1 (valid user D#)
        g0[1] = lds_off;                                      // lds_addr
        g0[2] = (unsigned)ga;                                 // global_addr[31:0]
        g0[3] = (unsigned)((ga >> 32) & 0x01FFFFFFu) | 0x80000000u; // ga[56:32] | type=2
        i32x8 g1;
        g1[0] = (int)(2u << 16);                              // data_size=4B, wg_mask=0
        g1[1] = (int)((unsigned)NN << 16);                    // tensor_dim0[15:0]
        g1[2] = (int)((unsigned)CC << 16);                    // tensor_dim0[31:16]=0 | tensor_dim1[15:0]
        g1[3] = (int)(16u << 16);                             // tensor_dim1[31:16]=0 | tile_dim0=16
        g1[4] = (int)CC;                                      // tile_dim1=192, tile_dim2=0
        g1[5] = (int)NN;                                      // tensor_dim0_stride[31:0]
        g1[6] = 0;                                            // stride0[47:32] | stride1[15:0]
        g1[7] = 0;
        i32x4 gz4 = {0, 0, 0, 0};                             // 2-D tensor: groups 2/3 unused
        i32x8 gz8 = {0, 0, 0, 0, 0, 0, 0, 0};                 // trailing operand: zero-filled
        __builtin_amdgcn_tensor_load_to_lds(g0, g1, gz4, gz4, gz8, 0);
        __builtin_amdgcn_s_wait_tensorcnt(0);
    }
#else
    for (int i = threadIdx.x; i < CC * 16; i += 256) {        // fallback: cooperative copy
        int c = i >> 4, j = i & 15;
        panel[i] = src[(size_t)c * NN + j];
    }
#endif
    __syncthreads();

    int wv = threadIdx.x >> 5;                  // 0..7
    int lane = threadIdx.x & 31;
    int r = lane & 15;
    int koff = (lane >> 4) * 2;
    int rowoff = 8 * (lane >> 4);
#pragma unroll 1
    for (int tt = 0; tt < MT_E / 8; ++tt) {     // 6 M-tiles per wave
        int m0 = (wv * (MT_E / 8) + tt) * 16;
        const float* abase = Wc + (size_t)(m0 + r) * CC + koff;
        v8f acc = {0.f, 0.f, 0.f, 0.f, 0.f, 0.f, 0.f, 0.f};
        for (int k = 0; k < CC; k += 4) {
            v2f a;  a.x  = abase[k]; a.y = abase[k + 1];
            v2f bv; bv.x = panel[(k + koff) * 16 + r];
            bv.y        = panel[(k + koff + 1) * 16 + r];
            acc = __builtin_amdgcn_wmma_f32_16x16x4_f32(false, a, false, bv, (short)0, acc, false, false);
        }
        float* outp = AB + ((size_t)(b * CTOT + m0 + rowoff)) * NN + n0 + r;
#pragma unroll
        for (int v = 0; v < 8; ++v) outp[(size_t)v * NN] = acc[v];
    }
}

// ---------------- K9: out[b,o,n] = relu( A[o,n] + b_edge[o] + max_k B[o, idx[n,k]] ) ----------------
__global__ void k_final(const float* __restrict__ AB, const int* __restrict__ idx,
                        const float* __restrict__ b_edge, float* __restrict__ out) {
    int t = blockIdx.x * blockDim.x + threadIdx.x;
    if (t >= BB * COUT * NN) return;
    int n = t % NN; int o = (t / NN) % COUT; int b = t / (NN * COUT);
    float a = AB[((size_t)(b * CTOT + o)) * NN + n] + b_edge[o];
    const float* brow = AB + ((size_t)(b * CTOT + COUT + o)) * NN;
    const int* irow = idx + ((size_t)(b * NN + n)) * KNN_;
    float best = -INFINITY;
#pragma unroll
    for (int j = 0; j < KNN_; ++j) best = fmaxf(best, brow[irow[j]]);
    out[t] = fmaxf(a + best, 0.f);
}

extern "C" void kernel_launch(void* const* d_in, const int* in_sizes, int n_in,
                              void* d_out, int out_size, void* d_ws, size_t ws_size,
                              hipStream_t stream) {
    const float* x      = (const float*)d_in[0];
    const float* w_edge = (const float*)d_in[1];
    const float* b_edge = (const float*)d_in[2];
    const float* w1     = (const float*)d_in[3];
    const float* b1     = (const float*)d_in[4];
    const float* w2     = (const float*)d_in[5];
    const float* b2     = (const float*)d_in[6];
    const float* w_sp   = (const float*)d_in[7];
    const float* b_sp   = (const float*)d_in[8];
    float* out = (float*)d_out;
    float* ws  = (float*)d_ws;

    // workspace layout (floats); AB aliases the dead xnT+G region (~60 MB total)
    size_t off = 0;
    float* pmean  = ws + off; off += (size_t)BB * CC;
    float* pmax   = ws + off; off += (size_t)BB * CC;
    float* ch     = ws + off; off += (size_t)BB * CC;
    float* sp_in  = ws + off; off += (size_t)BB * 2 * NN;
    float* sp_att = ws + off; off += (size_t)BB * NN;
    float* sq     = ws + off; off += (size_t)BB * NN;
    float* Wc     = ws + off; off += (size_t)CTOT * CC;
    float* xf_att = ws + off; off += (size_t)BB * CC * NN;
    float* xnT    = ws + off;                         // (B,N,C)
    float* AB     = xnT;                              // (B,768,N) reuses xnT+G after top-k
    off += (size_t)BB * NN * CC;
    float* G      = ws + off; off += (size_t)BB * NN * NN;
    int*   idxb   = (int*)(ws + off);                 // (B,N,9)

    const int tpb = 256;
    k_chan_pool<<<BB * CC, 256, 0, stream>>>(x, pmean, pmax);
    k_chan_mlp<<<BB, CC, 0, stream>>>(pmean, pmax, w1, b1, w2, b2, ch);
    k_sp_in  <<<(BB * NN + tpb - 1) / tpb, tpb, 0, stream>>>(x, ch, sp_in);
    k_sp_conv<<<(BB * NN + tpb - 1) / tpb, tpb, 0, stream>>>(sp_in, w_sp, b_sp, sp_att);
    k_att_row<<<BB * CC, 256, 0, stream>>>(x, ch, sp_att, xf_att);
    k_norm   <<<BB * NN, 256, 0, stream>>>(x, xnT, sq);
    k_gram_wmma<<<BB * MT_G * NT, 32, 0, stream>>>(xnT, G);
    k_topk   <<<(BB * NN + 127) / 128, 128, 0, stream>>>(G, sq, idxb);
    k_wc     <<<(CTOT * CC + tpb - 1) / tpb, tpb, 0, stream>>>(w_edge, Wc);
    k_edge_wmma<<<BB * NT, 256, 0, stream>>>(Wc, xf_att, AB);
    k_final  <<<(BB * COUT * NN + tpb - 1) / tpb, tpb, 0, stream>>>(AB, idxb, b_edge, out);
}